// LGN_Encoder_19344532701199
// MI455X (gfx1250) — compile-verified
//
#include <hip/hip_runtime.h>
#include <hip/hip_bf16.h>

#define NUSERS 200000
#define NITEMS 100000
#define NNODES 300000            // NUSERS + NITEMS
#define FDIM   64
#define NEDGE  10000000
#define ALPHA  0.25f             // 1/(N_LAYERS+1), N_LAYERS=3
#define BN_EPSF 1e-5f
#define SCAN_B 1024
#define NBLK   ((NNODES + SCAN_B - 1) / SCAN_B)   // 293

typedef long long i64;
typedef __attribute__((ext_vector_type(2))) int   v2i;
typedef __attribute__((ext_vector_type(2))) float v2f;
typedef __attribute__((ext_vector_type(4))) float v4f;
typedef __attribute__((ext_vector_type(8))) float v8f;

// ---------------------------------------------------------------------------
// K1: zero degree counters and BN stats accumulators
// ---------------------------------------------------------------------------
__global__ void k_init_scalars(unsigned* __restrict__ deg, float* __restrict__ stats) {
    int i = blockIdx.x * blockDim.x + threadIdx.x;
    if (i < NNODES) deg[i] = 0u;
    if (i < 128)    stats[i] = 0.0f;
}

// ---------------------------------------------------------------------------
// K2: degree of destination (col) nodes  (10M u32 atomics, once per call)
//     Edge stream read non-temporally (single use, don't pollute L2).
// ---------------------------------------------------------------------------
__global__ void k_degree(const i64* __restrict__ col, unsigned* __restrict__ deg) {
    int stride = gridDim.x * blockDim.x;
    for (int e = blockIdx.x * blockDim.x + threadIdx.x; e < NEDGE; e += stride) {
        __builtin_prefetch(&col[e + stride], 0, 1);
        i64 c = __builtin_nontemporal_load(&col[e]);
        atomicAdd(&deg[(int)c], 1u);
    }
}

// ---------------------------------------------------------------------------
// K3: dis[i] = deg>0 ? rsqrt(deg) : 0
// ---------------------------------------------------------------------------
__global__ void k_dis(const unsigned* __restrict__ deg, float* __restrict__ dis) {
    int i = blockIdx.x * blockDim.x + threadIdx.x;
    if (i < NNODES) {
        unsigned d = deg[i];
        dis[i] = (d > 0u) ? __frsqrt_rn((float)d) : 0.0f;
    }
}

// ---------------------------------------------------------------------------
// Exclusive prefix scan of deg -> off, in three stages.
// ---------------------------------------------------------------------------
__global__ void k_scan_local(const unsigned* __restrict__ deg, unsigned* __restrict__ off,
                             unsigned* __restrict__ bsum) {
    __shared__ unsigned tmp[SCAN_B];
    const int tid = threadIdx.x;
    const int i = blockIdx.x * SCAN_B + tid;
    unsigned v = (i < NNODES) ? deg[i] : 0u;
    tmp[tid] = v;
    __syncthreads();
    for (int d = 1; d < SCAN_B; d <<= 1) {
        unsigned t = (tid >= d) ? tmp[tid - d] : 0u;
        __syncthreads();
        tmp[tid] += t;
        __syncthreads();
    }
    if (i < NNODES) off[i] = tmp[tid] - v;            // exclusive within chunk
    if (tid == SCAN_B - 1) bsum[blockIdx.x] = tmp[tid];
}

__global__ void k_scan_bsum(unsigned* __restrict__ bsum) {   // single block, 512 thr
    __shared__ unsigned tmp[512];
    const int tid = threadIdx.x;
    unsigned v = (tid < NBLK) ? bsum[tid] : 0u;
    tmp[tid] = v;
    __syncthreads();
    for (int d = 1; d < 512; d <<= 1) {
        unsigned t = (tid >= d) ? tmp[tid - d] : 0u;
        __syncthreads();
        tmp[tid] += t;
        __syncthreads();
    }
    if (tid < NBLK) bsum[tid] = tmp[tid] - v;          // exclusive block offsets
}

__global__ void k_scan_add(unsigned* __restrict__ off, const unsigned* __restrict__ bsum,
                           unsigned* __restrict__ cursor) {
    int i = blockIdx.x * blockDim.x + threadIdx.x;
    if (i < NNODES) {
        unsigned o = off[i] + bsum[i >> 10];
        off[i]    = o;
        cursor[i] = o;                                 // fill-pass cursors
    }
    if (i == 0) off[NNODES] = NEDGE;                   // total degree is known
}

// ---------------------------------------------------------------------------
// K4: CSR fill. Packs (src, norm) into one int2 (single b64 store per edge).
//     norm computed on the fly from L2-resident dis (1.2MB). csr stores stay
//     RT so scattered 8B writes coalesce in L2.
// ---------------------------------------------------------------------------
__global__ void k_fill(const i64* __restrict__ row, const i64* __restrict__ col,
                       const float* __restrict__ dis,
                       unsigned* __restrict__ cursor, int2* __restrict__ csr) {
    int stride = gridDim.x * blockDim.x;
    for (int e = blockIdx.x * blockDim.x + threadIdx.x; e < NEDGE; e += stride) {
        __builtin_prefetch(&row[e + stride], 0, 1);
        __builtin_prefetch(&col[e + stride], 0, 1);
        int r = (int)__builtin_nontemporal_load(&row[e]);
        int c = (int)__builtin_nontemporal_load(&col[e]);
        float w = dis[r] * dis[c];
        unsigned pos = atomicAdd(&cursor[c], 1u);
        csr[pos] = make_int2(r, __float_as_int(w));
    }
}

// ---------------------------------------------------------------------------
// K5: acc = ALPHA * x   (acc lives in d_out; NT store — only re-read much later)
// ---------------------------------------------------------------------------
__global__ void k_init_acc(const float* __restrict__ x, float* __restrict__ acc) {
    int stride = gridDim.x * blockDim.x;
    const int n4 = NNODES * FDIM / 4;
    for (int i = blockIdx.x * blockDim.x + threadIdx.x; i < n4; i += stride) {
        v4f v = ((const v4f*)x)[i];                    // RT: x is layer-0 gather source
        v4f o = ALPHA * v;
        __builtin_nontemporal_store(o, &((v4f*)acc)[i]);
    }
}

// ---------------------------------------------------------------------------
// K6: CSR propagation, atomic-free. One wave32 per destination node; lane l
//     owns features [2l, 2l+1]. Neighbor records staged 32-at-a-time with one
//     coalesced NT b64 load, broadcast via forced v_readlane. The random
//     float2 gathers from hsrc stay RT (L2-resident, 76.8MB << 192MB L2);
//     hdst stores stay RT (next layer's gather source); acc RMW is NT.
//     Fuses acc += ALPHA * h_new.
// ---------------------------------------------------------------------------
__global__ void k_prop_csr(const unsigned* __restrict__ off, const int2* __restrict__ csr,
                           const float* __restrict__ hsrc, float* __restrict__ hdst,
                           float* __restrict__ acc) {
    const int lane   = threadIdx.x & 31;
    const int wave   = (blockIdx.x * blockDim.x + threadIdx.x) >> 5;
    const int nwaves = (gridDim.x * blockDim.x) >> 5;
    for (int v = wave; v < NNODES; v += nwaves) {
        const unsigned beg = off[v];
        const unsigned end = off[v + 1];
        float sx = 0.f, sy = 0.f;
        for (unsigned base = beg; base < end; base += 32u) {
            const int navail = (int)min(32u, end - base);
            v2i p; p.x = 0; p.y = 0;
            if (lane < navail)                          // coalesced NT b64
                p = __builtin_nontemporal_load((const v2i*)(csr + base + lane));
            for (int j = 0; j < navail; ++j) {
                int   src = __builtin_amdgcn_readlane(p.x, j);   // v_readlane_b32
                float w   = __int_as_float(__builtin_amdgcn_readlane(p.y, j));
                const float2 hv = *(const float2*)(hsrc + src * FDIM + lane * 2);
                sx = fmaf(w, hv.x, sx);
                sy = fmaf(w, hv.y, sy);
            }
        }
        float2* hd = (float2*)(hdst + v * FDIM + lane * 2);
        *hd = make_float2(sx, sy);                       // RT: next layer gathers this
        v2f* ap = (v2f*)(acc + v * FDIM + lane * 2);
        v2f a = __builtin_nontemporal_load(ap);
        a.x += ALPHA * sx;
        a.y += ALPHA * sy;
        __builtin_nontemporal_store(a, ap);
    }
}

// ---------------------------------------------------------------------------
// K8: BN column statistics via WMMA.  colsum = ones(16x4) x B(4x16) tiles,
//     accumulated in the f32 D registers of V_WMMA_F32_16X16X4_F32.
// ---------------------------------------------------------------------------
__global__ void k_stats(const float* __restrict__ acc, float* __restrict__ stats) {
    const int lane   = threadIdx.x & 31;
    const int wave   = (blockIdx.x * blockDim.x + threadIdx.x) >> 5;
    const int nwaves = (gridDim.x * blockDim.x) >> 5;   // multiple of 4 by launch config
    const int cb      = wave & 3;                       // column block: cols [16*cb, 16*cb+16)
    const int cstride = nwaves >> 2;

    v2f a; a.x = 1.0f; a.y = 1.0f;                      // A = ones (16x4)
    v8f csum = {};
    v8f csq  = {};
    const int n     = lane & 15;
    const int khalf = (lane >> 4) * 2;                  // 0 for lanes 0-15, 2 for 16-31

    for (int chunk = (wave >> 2); chunk < NNODES / 4; chunk += cstride) {
        const float* base = acc + (size_t)(chunk * 4) * FDIM + cb * 16 + n;
        v2f b, b2;
        b.x = base[(khalf + 0) * FDIM];
        b.y = base[(khalf + 1) * FDIM];
        b2.x = b.x * b.x;
        b2.y = b.y * b.y;
        csum = __builtin_amdgcn_wmma_f32_16x16x4_f32(false, a, false, b,  (short)0, csum, false, false);
        csq  = __builtin_amdgcn_wmma_f32_16x16x4_f32(false, a, false, b2, (short)0, csq,  false, false);
    }
    if (lane < 16) {
        __hip_atomic_fetch_add(&stats[cb * 16 + lane],      csum[0], __ATOMIC_RELAXED, __HIP_MEMORY_SCOPE_AGENT);
        __hip_atomic_fetch_add(&stats[64 + cb * 16 + lane], csq[0],  __ATOMIC_RELAXED, __HIP_MEMORY_SCOPE_AGENT);
    }
}

// ---------------------------------------------------------------------------
// K9: in-place batch-norm finalize:  out = (acc-mean)*rsqrt(var+eps)*w + b
// ---------------------------------------------------------------------------
__global__ void k_bn(float* __restrict__ out, const float* __restrict__ stats,
                     const float* __restrict__ w, const float* __restrict__ bias) {
    int stride = gridDim.x * blockDim.x;
    const float invN = 1.0f / (float)NNODES;
    for (int i = blockIdx.x * blockDim.x + threadIdx.x; i < NNODES * FDIM; i += stride) {
        int c = i & (FDIM - 1);
        float mean = stats[c] * invN;
        float var  = stats[64 + c] * invN - mean * mean;
        float s    = __frsqrt_rn(var + BN_EPSF) * w[c];
        float o    = (out[i] - mean) * s + bias[c];
        __builtin_nontemporal_store(o, &out[i]);       // final output, no reuse
    }
}

// ---------------------------------------------------------------------------
extern "C" void kernel_launch(void* const* d_in, const int* in_sizes, int n_in,
                              void* d_out, int out_size, void* d_ws, size_t ws_size,
                              hipStream_t stream) {
    const float* x   = (const float*)d_in[0];
    const i64*   row = (const i64*)d_in[1];          // edge_index[0], int64
    const i64*   col = row + NEDGE;                  // edge_index[1]
    const float* bnw = (const float*)d_in[2];
    const float* bnb = (const float*)d_in[3];

    // workspace carve-up (~240 MB); csr first for 8-byte alignment
    int2*     csr    = (int2*)d_ws;                              // NEDGE int2 (80MB)
    float*    h_a    = (float*)(csr + NEDGE);                    // NNODES*FDIM
    float*    h_b    = h_a + (size_t)NNODES * FDIM;              // NNODES*FDIM
    unsigned* deg    = (unsigned*)(h_b + (size_t)NNODES * FDIM); // NNODES (reused as cursor)
    float*    dis    = (float*)(deg + NNODES);                   // NNODES
    unsigned* off    = (unsigned*)(dis + NNODES);                // NNODES+1
    unsigned* bsum   = off + (NNODES + 1);                       // NBLK (pad 512)
    float*    stats  = (float*)(bsum + 512);                     // 128
    float*    acc    = (float*)d_out;                            // output accumulated in-place

    const int B = 256;
    k_init_scalars<<<(NNODES + B - 1) / B, B, 0, stream>>>(deg, stats);
    k_degree<<<2048, B, 0, stream>>>(col, deg);
    k_dis<<<(NNODES + B - 1) / B, B, 0, stream>>>(deg, dis);

    // exclusive scan deg -> off (and cursor copy in deg)
    k_scan_local<<<NBLK, SCAN_B, 0, stream>>>(deg, off, bsum);
    k_scan_bsum<<<1, 512, 0, stream>>>(bsum);
    k_scan_add<<<(NNODES + B - 1) / B, B, 0, stream>>>(off, bsum, deg);

    k_fill<<<2048, B, 0, stream>>>(row, col, dis, deg, csr);
    k_init_acc<<<2048, B, 0, stream>>>(x, acc);

    // 3 propagation layers, atomic-free CSR gather; x is the layer-0 source
    k_prop_csr<<<2048, B, 0, stream>>>(off, csr, x,   h_a, acc);
    k_prop_csr<<<2048, B, 0, stream>>>(off, csr, h_a, h_b, acc);
    k_prop_csr<<<2048, B, 0, stream>>>(off, csr, h_b, h_a, acc);

    // 512 blocks * 256 threads = 4096 waves (multiple of 4 column-block streams)
    k_stats<<<512, B, 0, stream>>>(acc, stats);
    k_bn<<<2048, B, 0, stream>>>(acc, stats, bnw, bnb);
}